// RandomEigenframeFilter_36369783062583
// MI455X (gfx1250) — compile-verified
//
#include <hip/hip_runtime.h>

// Problem constants (from reference): B=4, F=8 -> BF=32, N=4096, CIN=COUT=32, KM=256
#define BF   32
#define NN   4096
#define CIN  32
#define COUT 32
#define KM   256
#define INV_SQRT_M 0.17677669529663687f  // 1/sqrt(32)

typedef float v2f __attribute__((ext_vector_type(2)));
typedef float v8f __attribute__((ext_vector_type(8)));
typedef float f4  __attribute__((ext_vector_type(4)));

// D = A(16x4 f32) * B(4x16 f32) + C(16x16 f32)
#define WMMA_F32(a, b, c)                                                     \
  __builtin_amdgcn_wmma_f32_16x16x4_f32(false, (a), false, (b), (short)0,     \
                                        (c), false, false)

// ---------------------------------------------------------------------------
// K1: G[bf][m][i] = sum_n D[bf][n][m] * x[bf][n][i]
// grid = 256 blocks (32 bf x 8 m-groups of 32), block = 256 threads (8 waves)
// wave = (mt in {0,1}) x (ks in {0..3}); each wave: 16 m rows, K-chunk 1024.
// ---------------------------------------------------------------------------
__global__ __launch_bounds__(256) void k1_dtx(const float* __restrict__ x,
                                              const float* __restrict__ D,
                                              float* __restrict__ G) {
  const int bx   = blockIdx.x;
  const int bf   = bx >> 3;
  const int m0   = (bx & 7) * 32;
  const int tid  = threadIdx.x;
  const int lane = tid & 31;
  const int wave = tid >> 5;      // 0..7
  const int mt   = wave & 1;      // m-tile within group
  const int ks   = wave >> 1;     // K split 0..3
  const int l15  = lane & 15;
  const int kb   = (lane >> 4) << 1;  // 0 for lanes 0-15, 2 for lanes 16-31

  const float* Dbf = D + (size_t)bf * NN * KM;
  const float* xbf = x + (size_t)bf * NN * CIN;
  const int mbase  = m0 + mt * 16;
  const int k0     = ks * 1024;

  v8f acc0 = {};  // i = 0..15
  v8f acc1 = {};  // i = 16..31

  // A lane view: a.x = D[k+kb+0][mbase+l15], a.y = D[k+kb+1][mbase+l15]
  // B lane view: b.x = x[k+kb+0][i],        b.y = x[k+kb+1][i]
  const float* pA = Dbf + (size_t)(k0 + kb) * KM + mbase + l15;
  const float* pB = xbf + (size_t)(k0 + kb) * CIN + l15;

#pragma unroll 4
  for (int it = 0; it < 256; ++it) {  // 256 * 4 = 1024 K values
    v2f a, b0, b1;
    a.x  = pA[0];
    a.y  = pA[KM];
    b0.x = pB[0];
    b0.y = pB[CIN];
    b1.x = pB[16];
    b1.y = pB[CIN + 16];
    acc0 = WMMA_F32(a, b0, acc0);
    acc1 = WMMA_F32(a, b1, acc1);
    pA += 4 * KM;
    pB += 4 * CIN;
  }

  // Cross-wave reduction over the 4 K splits via LDS.
  __shared__ float red[8][2][256];  // [wave][i-tile][row*16+col] = 16KB
#pragma unroll
  for (int r = 0; r < 8; ++r) {
    const int row = r + ((lane >> 4) << 3);  // C layout: M = r + 8*(lane/16)
    red[wave][0][row * 16 + l15] = acc0[r];
    red[wave][1][row * 16 + l15] = acc1[r];
  }
  __syncthreads();

#pragma unroll
  for (int jj = 0; jj < 4; ++jj) {
    const int j     = tid + (jj << 8);  // 0..1023
    const int idx   = j & 255;
    const int itile = (j >> 8) & 1;
    const int mtile = j >> 9;
    // wave index = ks*2 + mtile
    const float s = red[0 + mtile][itile][idx] + red[2 + mtile][itile][idx] +
                    red[4 + mtile][itile][idx] + red[6 + mtile][itile][idx];
    const int m = m0 + mtile * 16 + (idx >> 4);
    const int i = itile * 16 + (idx & 15);
    G[((size_t)bf * KM + m) * CIN + i] = s;
  }
}

// ---------------------------------------------------------------------------
// K2: S[bf][m][o] = (1/sqrt(32)) * sum_i coeff[o][i][m] * G[bf][m][i]
// 262144 threads total; m contiguous across lanes -> coeff loads coalesced.
// ---------------------------------------------------------------------------
__global__ __launch_bounds__(256) void k2_scaled(const float* __restrict__ coeff,
                                                 const float* __restrict__ G,
                                                 float* __restrict__ S) {
  const int tid = blockIdx.x * 256 + threadIdx.x;  // 0..262143
  const int m   = tid & (KM - 1);
  const int o   = (tid >> 8) & (COUT - 1);
  const int bf  = tid >> 13;
  const float* g = G + ((size_t)bf * KM + m) * CIN;
  float s = 0.f;
#pragma unroll 8
  for (int i = 0; i < CIN; ++i)
    s += coeff[((size_t)o * CIN + i) * KM + m] * g[i];
  S[((size_t)bf * KM + m) * COUT + o] = s * INV_SQRT_M;
}

// ---------------------------------------------------------------------------
// K3: out[bf][n][o] = sum_m D[bf][n][m] * S[bf][m][o]
// grid = 1024 blocks (32 bf x 32 n-tiles of 128), block = 256 threads.
// S slice staged once in LDS (32KB); D tiles staged per 32-m chunk (pad 36
// floats/row => conflict-free column reads for the A operand).
// ---------------------------------------------------------------------------
__global__ __launch_bounds__(256) void k3_out(const float* __restrict__ D,
                                              const float* __restrict__ S,
                                              float* __restrict__ out) {
  const int bx   = blockIdx.x;
  const int bf   = bx >> 5;
  const int n0   = (bx & 31) * 128;
  const int tid  = threadIdx.x;
  const int lane = tid & 31;
  const int wave = tid >> 5;
  const int l15  = lane & 15;
  const int kb   = (lane >> 4) << 1;

  __shared__ float sS[KM * COUT];  // [m][o], 32KB
  __shared__ float sD[128 * 36];   // 128 rows, padded stride 36, 18KB

  // Stage S[bf] (256x32 f32) cooperatively with b128 loads.
  {
    const f4* S4  = (const f4*)(S + (size_t)bf * KM * COUT);
    f4*       sS4 = (f4*)sS;
#pragma unroll
    for (int j = 0; j < 8; ++j) sS4[tid + j * 256] = S4[tid + j * 256];
  }

  const float* Dbf = D + (size_t)bf * NN * KM;
  const int rowbase = wave * 16;  // 16 n rows per wave

  v8f acc0 = {};
  v8f acc1 = {};

  for (int mc = 0; mc < KM; mc += 32) {
    __syncthreads();  // protect sD from previous iteration's readers
    // Stage D[n0..n0+127][mc..mc+31]: 1024 float4, 4 per thread, coalesced.
#pragma unroll
    for (int j = 0; j < 4; ++j) {
      const int idx = tid + j * 256;  // float4 index
      const int row = idx >> 3;
      const int c4  = idx & 7;
      const f4 v = *(const f4*)(Dbf + (size_t)(n0 + row) * KM + mc + c4 * 4);
      float* dst = &sD[row * 36 + c4 * 4];
      dst[0] = v.x; dst[1] = v.y; dst[2] = v.z; dst[3] = v.w;
    }
    __syncthreads();  // also orders the sS stage before first use

#pragma unroll
    for (int kk = 0; kk < 32; kk += 4) {
      v2f a, b0, b1;
      const float* pa = &sD[(rowbase + l15) * 36 + kk + kb];
      a.x = pa[0];
      a.y = pa[1];
      const float* pb = &sS[(size_t)(mc + kk + kb) * COUT + l15];
      b0.x = pb[0];
      b0.y = pb[COUT];
      b1.x = pb[16];
      b1.y = pb[COUT + 16];
      acc0 = WMMA_F32(a, b0, acc0);
      acc1 = WMMA_F32(a, b1, acc1);
    }
  }

  // Write out: lane l, vgpr r -> row = r + 8*(l/16), col = l%16.
  float* obase = out + ((size_t)bf * NN + n0 + rowbase) * COUT;
#pragma unroll
  for (int r = 0; r < 8; ++r) {
    const int row = r + ((lane >> 4) << 3);
    obase[(size_t)row * COUT + l15]      = acc0[r];
    obase[(size_t)row * COUT + 16 + l15] = acc1[r];
  }
}

// ---------------------------------------------------------------------------
extern "C" void kernel_launch(void* const* d_in, const int* in_sizes, int n_in,
                              void* d_out, int out_size, void* d_ws,
                              size_t ws_size, hipStream_t stream) {
  const float* x     = (const float*)d_in[0];  // (4,8,4096,32)
  const float* D     = (const float*)d_in[1];  // (4,8,4096,256)
  const float* coeff = (const float*)d_in[2];  // (32,32,256)
  float* out = (float*)d_out;                  // (4,8,4096,32)

  float* G = (float*)d_ws;                     // 32*256*32 f32 = 1MB
  float* S = G + (size_t)BF * KM * CIN;        // 32*256*32 f32 = 1MB

  k1_dtx<<<dim3(BF * 8), dim3(256), 0, stream>>>(x, D, G);
  k2_scaled<<<dim3((BF * KM * COUT) / 256), dim3(256), 0, stream>>>(coeff, G, S);
  k3_out<<<dim3(BF * 32), dim3(256), 0, stream>>>(D, S, out);
}